// GCNLayer_34411277975785
// MI455X (gfx1250) — compile-verified
//
#include <hip/hip_runtime.h>
#include <hip/hip_bf16.h>

typedef __attribute__((ext_vector_type(2))) float v2f;
typedef __attribute__((ext_vector_type(8))) float v8f;

#define NN 50000
#define NE 800000
#define DD 128

// ---------------- zero workspace ----------------
__global__ void k_zero(float* __restrict__ p, int n) {
    int t = blockIdx.x * blockDim.x + threadIdx.x;
    if (t < n) p[t] = 0.0f;
}

// ---------------- degree count ----------------
__global__ void k_deg(const int* __restrict__ dst, float* __restrict__ deg, int ne) {
    int t = blockIdx.x * blockDim.x + threadIdx.x;
    if (t < ne) atomicAdd(&deg[dst[t]], 1.0f);
}

// ---------------- norm = clip(deg,1)^-0.5 (in place) ----------------
__global__ void k_norm(float* __restrict__ deg, int nn) {
    int t = blockIdx.x * blockDim.x + threadIdx.x;
    if (t < nn) deg[t] = rsqrtf(fmaxf(deg[t], 1.0f));
}

// ---------------- scatter: agg[dst] += feat[src]*norm[src] ----------------
// one wave (32 lanes) per edge; each lane handles 4 consecutive floats (float4)
__global__ void k_scatter(const float* __restrict__ feat, const float* __restrict__ norm,
                          const int* __restrict__ src, const int* __restrict__ dst,
                          float* __restrict__ agg, int ne) {
    int t = blockIdx.x * blockDim.x + threadIdx.x;
    int e = t >> 5;
    if (e >= ne) return;
    int lane = t & 31;
    int s = src[e];
    int d = dst[e];
    float nm = norm[s];
    const float4* f4 = (const float4*)(feat + (size_t)s * DD);
    float4 v = f4[lane];
    float* out = agg + (size_t)d * DD + lane * 4;
    atomicAdd(out + 0, v.x * nm);
    atomicAdd(out + 1, v.y * nm);
    atomicAdd(out + 2, v.z * nm);
    atomicAdd(out + 3, v.w * nm);
}

// ---------------- GEMM via V_WMMA_F32_16X16X4_F32 ----------------
// block = 16 output rows (nodes); 8 waves cover the 8 n-tiles of D=128.
// h2[m][n] = (sum_k agg[m][k]*W[n][k] + b[n]) * norm[m], written in place over agg.
__global__ void __launch_bounds__(256)
k_gemm_wmma(const float* __restrict__ W, const float* __restrict__ bias,
            const float* __restrict__ norm, float* __restrict__ aggh) {
    const int mbase = blockIdx.x * 16;
    const int wave  = threadIdx.x >> 5;   // 0..7 -> n-tile
    const int lane  = threadIdx.x & 31;
    const int nbase = wave * 16;
    const int mn    = lane & 15;          // row (for A) / col (for B) within tile
    const int khalf = (lane >> 4) << 1;   // lanes 0-15 hold K={0,1}, lanes 16-31 K={2,3}

    const float* arow = aggh + (size_t)(mbase + mn) * DD;   // A: agg row
    const float* brow = W    + (size_t)(nbase + mn) * DD;   // B: column n of W^T = row n of W

    v8f c = {0.f, 0.f, 0.f, 0.f, 0.f, 0.f, 0.f, 0.f};
    #pragma unroll 4
    for (int k0 = 0; k0 < DD; k0 += 4) {
        const v2f a = *(const v2f*)(arow + k0 + khalf);   // 8B-aligned
        const v2f b = *(const v2f*)(brow + k0 + khalf);
        c = __builtin_amdgcn_wmma_f32_16x16x4_f32(
                /*neg_a=*/false, a, /*neg_b=*/false, b,
                /*c_mod=*/(short)0, c, /*reuse_a=*/false, /*reuse_b=*/false);
    }

    __syncthreads();   // all 8 waves finished reading agg rows [mbase, mbase+16)

    // C/D layout: VGPR r -> M = r + 8*(lane>>4), N = lane&15
    const int n  = nbase + mn;
    const float bv = bias[n];
    const int mhi = (lane >> 4) * 8;
    #pragma unroll
    for (int r = 0; r < 8; ++r) {
        const int m = mbase + mhi + r;
        aggh[(size_t)m * DD + n] = (c[r] + bv) * norm[m];
    }
}

// ---------------- BN stats: per-column sum & sumsq ----------------
// 128 threads/block: thread j owns column j over a row-chunk; atomic reduce.
__global__ void k_bnstats(const float* __restrict__ h2, float* __restrict__ sums, int nn) {
    const int j = threadIdx.x;                       // 0..127
    const int rpb = (nn + gridDim.x - 1) / gridDim.x;
    const int r0 = blockIdx.x * rpb;
    const int r1 = min(r0 + rpb, nn);
    float s = 0.0f, ss = 0.0f;
    for (int r = r0; r < r1; ++r) {
        const float v = h2[(size_t)r * DD + j];
        s += v;
        ss += v * v;
    }
    atomicAdd(&sums[j], s);
    atomicAdd(&sums[DD + j], ss);
}

// ---------------- final: BN affine + ReLU + residual ----------------
__global__ void k_out(const float* __restrict__ feat, const float* __restrict__ h2,
                      const float* __restrict__ sums, const float* __restrict__ gamma,
                      const float* __restrict__ beta, float* __restrict__ out, int total) {
    int t = blockIdx.x * blockDim.x + threadIdx.x;
    if (t >= total) return;
    const int j = t & (DD - 1);
    const float inv_n = 1.0f / (float)NN;
    const float mean = sums[j] * inv_n;
    const float var  = sums[DD + j] * inv_n - mean * mean;
    const float sc   = gamma[j] * rsqrtf(var + 1e-5f);
    const float sh   = beta[j] - mean * sc;
    const float h    = h2[t] * sc + sh;
    out[t] = feat[t] + fmaxf(h, 0.0f);
}

extern "C" void kernel_launch(void* const* d_in, const int* in_sizes, int n_in,
                              void* d_out, int out_size, void* d_ws, size_t ws_size,
                              hipStream_t stream) {
    const float* feat  = (const float*)d_in[0];   // [50000,128]
    const float* W     = (const float*)d_in[1];   // [128,128]
    const float* bias  = (const float*)d_in[2];   // [128]
    const float* gamma = (const float*)d_in[3];   // [128]
    const float* beta  = (const float*)d_in[4];   // [128]
    const int* esrc    = (const int*)d_in[5];     // [800000]
    const int* edst    = (const int*)d_in[6];     // [800000]
    float* out = (float*)d_out;

    // workspace layout (floats)
    float* ws   = (float*)d_ws;
    float* deg  = ws;              // 50000 (becomes norm in place)
    float* sums = ws + 50176;      // 256 (sum | sumsq)
    float* agg  = ws + 50688;      // 6,400,000 (becomes h2 in place)
    const int ws_floats = 50688 + NN * DD;

    const int ne = in_sizes[5];
    const int total = NN * DD;

    // 1) zero deg + sums + agg (must re-zero every call)
    k_zero<<<(ws_floats + 255) / 256, 256, 0, stream>>>(ws, ws_floats);
    // 2) degree histogram
    k_deg<<<(ne + 255) / 256, 256, 0, stream>>>(edst, deg, ne);
    // 3) norm = rsqrt(max(deg,1))
    k_norm<<<(NN + 255) / 256, 256, 0, stream>>>(deg, NN);
    // 4) edge scatter: one wave per edge
    k_scatter<<<(ne * 32 + 255) / 256, 256, 0, stream>>>(feat, deg, esrc, edst, agg, ne);
    // 5) WMMA GEMM + bias + norm scale (h2 in place over agg); 50000/16 = 3125 blocks
    k_gemm_wmma<<<NN / 16, 256, 0, stream>>>(W, bias, deg, agg);
    // 6) BN statistics
    k_bnstats<<<256, DD, 0, stream>>>(agg, sums, NN);
    // 7) BN affine + ReLU + residual
    k_out<<<(total + 255) / 256, 256, 0, stream>>>(feat, agg, sums, gamma, beta, out, total);
}